// EdgeAwareMultiHeadAttention_6339371729566
// MI455X (gfx1250) — compile-verified
//
#include <hip/hip_runtime.h>
#include <hip/hip_bf16.h>
#include <math.h>

// Problem constants (match reference)
#define B_    4
#define N_    256
#define HID_  256
#define E_    64
#define H_    4
#define OUT_  128
#define D_    32
#define AGG_  260

// LDS layout (floats). E rows padded to stride 68 -> conflict-free b64 reads,
// 16B-aligned rows (68*4 = 272 = 16*17). TDM reproduces this padding in HW:
// pad_interval=5 (64 DWORDs) + pad_amount=3 (4 DWORDs).
#define SE_STRIDE 68
#define OFF_E     0
#define OFF_Q     (N_ * SE_STRIDE)        // 17408 floats = 69632 B (just past sE)
#define OFF_QK    (OFF_Q + OUT_)          // 17536
#define OFF_ATTN  (OFF_QK + H_ * E_)      // 17792
#define OFF_WE    (OFF_ATTN + H_ * N_)    // 18816
#define OFF_ASUM  (OFF_WE + H_ * E_)      // 19072
#define OFF_MEAN  (OFF_ASUM + 8)          // 19080
#define OFF_MAX   (OFF_MEAN + OUT_)       // 19208
#define SMEM_FLOATS (OFF_MAX + OUT_)      // 19336 floats = 77344 bytes

typedef float v2f __attribute__((ext_vector_type(2)));
typedef float v8f __attribute__((ext_vector_type(8)));
typedef unsigned int u32x4 __attribute__((ext_vector_type(4)));
typedef int i32x4 __attribute__((ext_vector_type(4)));
typedef int i32x8 __attribute__((ext_vector_type(8)));

__global__ __launch_bounds__(256)
void edge_attn_kernel(const float* __restrict__ h_x,
                      const float* __restrict__ h_e,
                      const float* __restrict__ h_m,
                      const float* __restrict__ W_Q,
                      const float* __restrict__ W_K,
                      const float* __restrict__ W_V,
                      const float* __restrict__ W_R,
                      float* __restrict__ out)
{
    extern __shared__ float smem[];
    float* sE    = smem + OFF_E;     // [256][68] edge row E = h_e[b,n,:,:]
    float* sQ    = smem + OFF_Q;     // [128]  q = h_x[b,n,:] @ W_Q^T
    float* sQK   = smem + OFF_QK;    // [4][64] qk[h,e] = sum_d q[h*32+d]*W_K[h*32+d,e]
    float* sAttn = smem + OFF_ATTN;  // [4][256] softmax over heads
    float* sWE   = smem + OFF_WE;    // [4][64] sum_m attn[h,m]*E[m,e]
    float* sASum = smem + OFF_ASUM;  // [4] sum_m attn[h,m]
    float* sMean = smem + OFF_MEAN;  // [128]
    float* sMax  = smem + OFF_MAX;   // [128]

    const int t  = threadIdx.x;
    const int bn = blockIdx.x;       // b*N + n

    // ---- Phase 0: stage E_row (256x64 f32 = 64KB) into LDS ----
#if __has_builtin(__builtin_amdgcn_tensor_load_to_lds)
    // Tensor Data Mover: 2D tile 256x64 of 4B elements, row-major, with the
    // 4-DWORD-per-row LDS padding generated by the TDM itself. Issued by wave 0
    // only (EXEC-independent, per-wave op); overlaps with Phase Q below.
    if (t < 32) {
        const unsigned long long ga =
            (unsigned long long)(const void*)h_e +
            (unsigned long long)bn * (unsigned long long)(N_ * E_ * 4);
        // D# group 0 (ISA 8.3): count=1 | lds_addr | global_addr | type=2
        u32x4 g0;
        g0[0] = 1u;                                         // count=1 (valid, user)
        g0[1] = 0u;                                         // lds_addr: sE at LDS base
        g0[2] = (unsigned int)(ga & 0xFFFFFFFFull);         // global_addr[31:0]
        g0[3] = (unsigned int)((ga >> 32) & 0x01FFFFFFull)  // global_addr[56:32]
                | (2u << 30);                               // type=2 ("image")
        // D# group 1 (ISA 8.4)
        i32x8 g1;
        g1[0] = (int)((2u << 16)      // data_size = 2 -> 4 bytes
                    | (1u << 20)      // pad_enable
                    | (5u << 22)      // pad_interval = 5 -> every 64 DWORDs
                    | (3u << 25));    // pad_amount   = 3 -> 4 DWORDs
        g1[1] = (int)(64u << 16);     // tensor_dim0 = 64  (lo16; abar addr = 0)
        g1[2] = (int)(256u << 16);    // tensor_dim1 = 256 (lo16; dim0 hi16 = 0)
        g1[3] = (int)(64u << 16);     // tile_dim0 = 64    (dim1 hi16 = 0)
        g1[4] = 0x100;                // tile_dim1 = 256, tile_dim2 = 0 (2D)
        g1[5] = 64;                   // tensor_dim0_stride = 64 elements
        g1[6] = 0;                    // dim0_stride hi / dim1_stride lo
        g1[7] = 0;                    // dim1_stride hi (unused, 2D)
        i32x4 gz = {0, 0, 0, 0};      // groups 2/3: zero (dims beyond 2D unused)
#if defined(__clang_major__) && __clang_major__ >= 23
        i32x8 gz8 = {0, 0, 0, 0, 0, 0, 0, 0};
        __builtin_amdgcn_tensor_load_to_lds(g0, g1, gz, gz, gz8, 0);
#else
        __builtin_amdgcn_tensor_load_to_lds(g0, g1, gz, gz, 0);
#endif
    }
#else
    // Fallback: float4 streaming copy through VGPRs
    {
        const float4* src = (const float4*)(h_e + (size_t)bn * (N_ * E_));
        for (int f = t; f < (N_ * E_) / 4; f += 256) {
            const int m = f >> 4;
            const int j = f & 15;
            float4 v = src[f];
            *(float4*)(sE + m * SE_STRIDE + 4 * j) = v;
        }
    }
#endif

    // ---- Phase Q: q[c] = h_x[b,n,:] . W_Q[c,:]  (global-only; overlaps TDM) ----
    if (t < OUT_) {
        __builtin_prefetch(W_R + (size_t)t * AGG_, 0, 0);  // warm L2 for final GEMV
        const float* xrow = h_x + (size_t)bn * HID_;
        const float* wq   = W_Q + (size_t)t * HID_;
        float acc = 0.0f;
        #pragma unroll 4
        for (int k = 0; k < HID_; ++k) acc += xrow[k] * wq[k];
        sQ[t] = acc;
    }
#if __has_builtin(__builtin_amdgcn_tensor_load_to_lds)
    if (t < 32) __builtin_amdgcn_s_wait_tensorcnt(0);  // wave 0 drains TDM
#endif
    __syncthreads();                                    // publishes sE + sQ to all

    // ---- Phase QK: qk[h,e] (folds K-projection into a [4x64] vector) ----
    {
        const int h = t >> 6, e = t & 63;
        float acc = 0.0f;
        #pragma unroll 4
        for (int d = 0; d < D_; ++d)
            acc += sQ[h * D_ + d] * W_K[(size_t)(h * D_ + d) * E_ + e];
        sQK[h * E_ + e] = acc;
    }
    __syncthreads();

    // ---- Phase scores + softmax over HEAD axis (axis=1, faithful) ----
    {
        const int m = t;
        const float scale = 0.17677669529663687f;  // 1/sqrt(32)
        float s[H_];
        const float* er = sE + m * SE_STRIDE;
        #pragma unroll
        for (int h = 0; h < H_; ++h) {
            const float* qk = sQK + h * E_;
            float acc = 0.0f;
            #pragma unroll 4
            for (int e = 0; e < E_; ++e) acc += er[e] * qk[e];
            s[h] = acc * scale;
        }
        if (h_m[bn] == 0.0f) {
            #pragma unroll
            for (int h = 0; h < H_; ++h) s[h] = -INFINITY;
        }
        float mx = fmaxf(fmaxf(s[0], s[1]), fmaxf(s[2], s[3]));
        float ex[H_], den = 0.0f;
        #pragma unroll
        for (int h = 0; h < H_; ++h) { ex[h] = __expf(s[h] - mx); den += ex[h]; }
        const float inv = 1.0f / den;
        #pragma unroll
        for (int h = 0; h < H_; ++h) sAttn[h * N_ + m] = ex[h] * inv;
    }
    __syncthreads();

    // ---- Phase WE: we_sum[h,e] = sum_m attn[h,m]*E[m,e]; attn_sum[h] ----
    {
        const int h = t >> 6, e = t & 63;
        const float* at = sAttn + h * N_;
        float acc = 0.0f, asum = 0.0f;
        for (int m = 0; m < N_; ++m) {
            const float a = at[m];
            acc  += a * sE[m * SE_STRIDE + e];
            asum += a;
        }
        sWE[h * E_ + e] = acc;
        if (e == 0) sASum[h] = asum;   // deterministic serial sum
    }
    __syncthreads();

    // ---- Phase mean: aggr_mean[c] = (we_sum[h,:] . W_V[c,:]) / (asum+1e-8) ----
    if (t < OUT_) {
        const int h = t >> 5;
        const float* wv = W_V + (size_t)t * E_;
        const float* we = sWE + h * E_;
        float acc = 0.0f;
        #pragma unroll 4
        for (int e = 0; e < E_; ++e) acc += we[e] * wv[e];
        sMean[t] = acc / (sASum[h] + 1e-8f);
    }
    // no extra barrier needed: WMMA phase only reads sE (phase 0) and sAttn (synced)

    // ---- Phase V-GEMM (WMMA f32 16x16x4) + weighted max over m ----
    // wave w owns output columns [16w,16w+16); 16 m-tiles; K-loop of 16 WMMAs.
    {
        const int lane = t & 31;
        const int wave = t >> 5;
        const int lid  = lane & 15;
        const int hi   = lane >> 4;          // K half: lanes 0-15 k={0,1}, 16-31 k={2,3}
        const int c    = wave * 16 + lid;    // output column 0..127
        const int hh   = (wave * 16) >> 5;   // head index (constant per wave)

        // B fragments: B[k,n] = W_V[c0+n][k], reused across all m-tiles
        v2f bf[16];
        #pragma unroll
        for (int ke = 0; ke < 16; ++ke) {
            const int klo = 4 * ke + 2 * hi;
            bf[ke] = *(const v2f*)(W_V + (size_t)c * E_ + klo);   // 8B aligned
        }

        float vmax = -INFINITY;
        for (int m0 = 0; m0 < N_; m0 += 16) {
            v8f acc = {};
            #pragma unroll
            for (int ke = 0; ke < 16; ++ke) {
                const int klo = 4 * ke + 2 * hi;
                const v2f a = *(const v2f*)(sE + (m0 + lid) * SE_STRIDE + klo);
                // D = A(16x4) x B(4x16) + C ; 8 args: neg_a,A,neg_b,B,c_mod,C,reuse_a,reuse_b
                acc = __builtin_amdgcn_wmma_f32_16x16x4_f32(
                          false, a, false, bf[ke], (short)0, acc, false, false);
            }
            // C layout: lane holds column c; VGPR r holds row m0 + r + 8*hi
            #pragma unroll
            for (int r = 0; r < 8; ++r) {
                const int m = m0 + r + 8 * hi;
                vmax = fmaxf(vmax, sAttn[hh * N_ + m] * acc[r]);
            }
        }
        // merge the two row-halves (lane L vs L+16 share the same column)
        vmax = fmaxf(vmax, __shfl_xor(vmax, 16, 32));
        if (hi == 0) sMax[c] = vmax;
    }
    __syncthreads();

    // ---- Final: out[o] = multi[B,N,260] . W_R[o,:] ----
    // multi layout per head h: [mean(32) | attn_sum(1) | max(32)] at a = h*65 + j
    if (t < OUT_) {
        const float* wr = W_R + (size_t)t * AGG_;
        float acc = 0.0f;
        #pragma unroll
        for (int h = 0; h < H_; ++h) {
            const float* w    = wr + h * (2 * D_ + 1);
            const float* mean = sMean + h * D_;
            const float* mmax = sMax  + h * D_;
            #pragma unroll 4
            for (int j = 0; j < D_; ++j) acc += mean[j] * w[j];
            acc += (sASum[h] + 1e-8f) * w[D_];
            #pragma unroll 4
            for (int j = 0; j < D_; ++j) acc += mmax[j] * w[D_ + 1 + j];
        }
        out[(size_t)bn * OUT_ + t] = acc;
    }
}

extern "C" void kernel_launch(void* const* d_in, const int* in_sizes, int n_in,
                              void* d_out, int out_size, void* d_ws, size_t ws_size,
                              hipStream_t stream) {
    const float* h_x = (const float*)d_in[0];
    const float* h_e = (const float*)d_in[1];
    const float* h_m = (const float*)d_in[2];
    const float* W_Q = (const float*)d_in[3];
    const float* W_K = (const float*)d_in[4];
    const float* W_V = (const float*)d_in[5];
    const float* W_R = (const float*)d_in[6];
    float* outp = (float*)d_out;

    const size_t smem = (size_t)SMEM_FLOATS * sizeof(float);  // ~77.3 KB < 320 KB/WGP
    edge_attn_kernel<<<B_ * N_, 256, smem, stream>>>(h_x, h_e, h_m, W_Q, W_K, W_V, W_R, outp);
}